// MulModule_25606595018768
// MI455X (gfx1250) — compile-verified
//
#include <hip/hip_runtime.h>

// ---------------------------------------------------------------------------
// neural VM step: out = x  (streamed copy)  + one-hot fixup at cols 80..111
// B = 262144 rows, D = 512 cols, fp32.  Memory-bound: ~1.07 GB @ 23.3 TB/s
// => ~46 us floor; compute is ~30 int/VALU ops per 2KB row.  Strategy:
// CDNA5 async global->LDS->global streaming (ASYNCcnt, b128, clause-issued),
// decode done in LDS between the copies.  Both global streams are touched
// exactly once => non-temporal (TH_NT, cpol=1) on loads and stores so the
// 1 GB stream does not churn the 192 MB L2.
// ---------------------------------------------------------------------------

#define DCOLS 512
#define ROWS_PER_TILE 16
#define TILE_ELEMS (ROWS_PER_TILE * DCOLS)              // 8192 floats = 32 KB
#define THREADS 256
#define CHUNKS_PER_THREAD (TILE_ELEMS / 4 / THREADS)    // 8 x 16B per thread

#define CPOL_TH_NT 1   // cpol[2:0] = TH; 1 = non-temporal (ISA 7.3)

typedef int v4i __attribute__((vector_size(16)));
typedef __attribute__((address_space(1))) v4i glob_v4i;  // HIP global AS
typedef __attribute__((address_space(3))) v4i lds_v4i;   // HIP shared AS

__device__ __forceinline__ void async_load_b128(const float* g, float* l) {
#if __has_builtin(__builtin_amdgcn_global_load_async_to_lds_b128)
    __builtin_amdgcn_global_load_async_to_lds_b128(
        (glob_v4i*)g, (lds_v4i*)l, 0, CPOL_TH_NT);
#else
    unsigned loff = (unsigned)(unsigned long long)l;   // low 32 bits = LDS offset
    asm volatile("global_load_async_to_lds_b128 %0, %1, off th:TH_LOAD_NT"
                 :: "v"(loff), "v"(g) : "memory");
#endif
}

__device__ __forceinline__ void async_store_b128(float* g, const float* l) {
#if __has_builtin(__builtin_amdgcn_global_store_async_from_lds_b128)
    __builtin_amdgcn_global_store_async_from_lds_b128(
        (glob_v4i*)g, (lds_v4i*)l, 0, CPOL_TH_NT);
#else
    unsigned loff = (unsigned)(unsigned long long)l;
    asm volatile("global_store_async_from_lds_b128 %0, %1, off th:TH_STORE_NT"
                 :: "v"(g), "v"(loff) : "memory");
#endif
}

__device__ __forceinline__ void wait_async0() {
#if __has_builtin(__builtin_amdgcn_s_wait_asynccnt)
    __builtin_amdgcn_s_wait_asynccnt(0);
#else
    asm volatile("s_wait_asynccnt 0" ::: "memory");
#endif
}

__global__ __launch_bounds__(THREADS)
void vm_step_kernel(const float* __restrict__ x, float* __restrict__ out) {
    __shared__ float tile[TILE_ELEMS];

    const int    t         = threadIdx.x;
    const size_t tile_base = (size_t)blockIdx.x * TILE_ELEMS;
    const float* gsrc = x   + tile_base;
    float*       gdst = out + tile_base;

    // ---- Phase 1: async stream tile global -> LDS (coalesced b128, NT) ----
#pragma unroll
    for (int k = 0; k < CHUNKS_PER_THREAD; ++k) {
        const int c = t + k * THREADS;            // 16-byte chunk index
        async_load_b128(gsrc + c * 4, &tile[c * 4]);
    }
    wait_async0();
    __syncthreads();

    // ---- Phase 2: decode + fixup, one wave32 per row -----------------------
    // Lane L owns cols [16L, 16L+16): each one-hot nibble bank sits entirely
    // in one lane (lane1=a_lo, lane2=a_hi, lane3=b_lo, lane4=b_hi, lane0=flags).
    const int wave = t >> 5;        // 0..7
    const int lane = t & 31;
#pragma unroll
    for (int rr = 0; rr < ROWS_PER_TILE / 8; ++rr) {
        const int row = wave * (ROWS_PER_TILE / 8) + rr;     // 0..15
        float* rbase = &tile[row * DCOLS];

        int bi = 0, act = 0;
        if (lane < 8) {
            float v[16];
#pragma unroll
            for (int j = 0; j < 4; ++j)
                ((float4*)v)[j] = ((const float4*)(rbase + lane * 16))[j];
            float bv = v[0];
#pragma unroll
            for (int i = 1; i < 16; ++i)          // strict '>' keeps FIRST max,
                if (v[i] > bv) { bv = v[i]; bi = i; }  // matching jnp.argmax
            act = (v[0] > 0.5f) && (v[1] > 0.5f);      // meaningful on lane 0
        }
        const int a_lo = __shfl(bi, 1, 32);
        const int a_hi = __shfl(bi, 2, 32);
        const int b_lo = __shfl(bi, 3, 32);
        const int b_hi = __shfl(bi, 4, 32);
        const int actv = __shfl(act, 0, 32);
        if (actv && lane == 0) {
            // exact integer equivalent of the fp32 magic-floor mod chain
            const int r8 = ((a_lo + 16 * a_hi) * (b_lo + 16 * b_hi)) & 255;
            rbase[80 + (r8 & 15)] += 1.0f;        // OUTPUT_LO one-hot
            rbase[96 + (r8 >> 4)] += 1.0f;        // OUTPUT_HI one-hot
        }
    }
    __syncthreads();

    // ---- Phase 3: async stream tile LDS -> global (NT) ---------------------
#pragma unroll
    for (int k = 0; k < CHUNKS_PER_THREAD; ++k) {
        const int c = t + k * THREADS;
        async_store_b128(gdst + c * 4, &tile[c * 4]);
    }
    wait_async0();
}

extern "C" void kernel_launch(void* const* d_in, const int* in_sizes, int n_in,
                              void* d_out, int out_size, void* d_ws, size_t ws_size,
                              hipStream_t stream) {
    (void)in_sizes; (void)n_in; (void)d_ws; (void)ws_size;
    const float* x   = (const float*)d_in[0];
    float*       out = (float*)d_out;

    const int nrows  = out_size / DCOLS;          // 262144
    const int blocks = nrows / ROWS_PER_TILE;     // 16384 (exact for this shape)
    vm_step_kernel<<<blocks, THREADS, 0, stream>>>(x, out);
}